// SplatterPhongShader_23081154249378
// MI455X (gfx1250) — compile-verified
//
#include <hip/hip_runtime.h>
#include <hip/hip_bf16.h>
#include <stdint.h>

// Problem constants (match reference)
#define NN   4
#define HH   256
#define WW   256
#define KK   4
#define TILE 16
#define SRC  18            // TILE + 1-pixel halo each side
#define NSRC (SRC * SRC)   // 324 source pixels per tile
#define BIGF 1.0e10f

// ---- gfx1250 async global->LDS DMA (CDNA5 ISA ch.10 / 15.18, ASYNCcnt) ----
__device__ __forceinline__ void async_load_b128(uint32_t lds_byte_addr, const void* gaddr) {
#if defined(__gfx1250__)
    asm volatile("global_load_async_to_lds_b128 %0, %1, off"
                 :: "v"(lds_byte_addr), "v"((unsigned long long)(uintptr_t)gaddr)
                 : "memory");
#endif
}
__device__ __forceinline__ void wait_asynccnt0() {
#if defined(__gfx1250__)
    asm volatile("s_wait_asynccnt 0" ::: "memory");
#endif
}

__global__ __launch_bounds__(256)
void splat_phong_kernel(const float* __restrict__ texels,
                        const float* __restrict__ normals,
                        const float* __restrict__ pcc,     // pixel_coords_cameras
                        const float* __restrict__ pcs,     // pixel_coords_screen (y,x)
                        const float* __restrict__ qdepth,
                        const unsigned char* __restrict__ bmask,
                        const float* __restrict__ lpos,
                        const float* __restrict__ lamb,
                        const float* __restrict__ ldif,
                        const float* __restrict__ lspc,
                        const float* __restrict__ cpos,
                        const float* __restrict__ mamb,
                        const float* __restrict__ mdif,
                        const float* __restrict__ mspc,
                        float* __restrict__ out)
{
    // LDS (all float4-aligned):
    //  sdep4[324]       : 4 layer depths per source pixel
    //  scrd4[324*2]     : screen coords, (y,x) pairs for layers {0,1} | {2,3}
    //  srgb4[324*4]     : shaded rgb per (pixel,layer), .w unused
    //  smask[324]       : 4-bit foreground mask per source pixel
    __shared__ alignas(16) float smem[NSRC*4 + NSRC*8 + NSRC*16 + NSRC];
    float4*   sdep4 = reinterpret_cast<float4*>(smem);
    float4*   scrd4 = reinterpret_cast<float4*>(smem + NSRC*4);
    float4*   srgb4 = reinterpret_cast<float4*>(smem + NSRC*12);
    unsigned* smask = reinterpret_cast<unsigned*>(smem + NSRC*28);

    const int tid = threadIdx.y * TILE + threadIdx.x;
    const int bx = blockIdx.x, by = blockIdx.y, n = blockIdx.z;
    const int gx0 = bx * TILE - 1, gy0 = by * TILE - 1;

    // ---- stage 1a: depth tile via async DMA (one b128 = 4 layer depths / pixel) ----
    for (int p = tid; p < NSRC; p += 256) {
        const int r = p / SRC, c = p % SRC;
        const int gy = gy0 + r, gx = gx0 + c;
        if ((unsigned)gy < HH && (unsigned)gx < WW) {
            const float* g = qdepth + ((((size_t)n*HH + gy)*WW) + gx) * KK;
            async_load_b128((uint32_t)(uintptr_t)(sdep4 + p), g);
        } else {
            sdep4[p] = make_float4(BIGF, BIGF, BIGF, BIGF);
        }
    }
    // ---- stage 1b: screen-coord tile via async DMA (2 x b128 / pixel) ----
    for (int c2 = tid; c2 < NSRC*2; c2 += 256) {
        const int p = c2 >> 1, half = c2 & 1;
        const int r = p / SRC, c = p % SRC;
        const int gy = gy0 + r, gx = gx0 + c;
        if ((unsigned)gy < HH && (unsigned)gx < WW) {
            const float* g = pcs + ((((((size_t)n*HH + gy)*WW) + gx)*KK) + half*2) * 2;
            async_load_b128((uint32_t)(uintptr_t)(scrd4 + p*2 + half), g);
        }
        // OOB: left uninitialized; mask-bit guard below guarantees it is never read
    }

    // ---- stage 2: 4-bit foreground mask per source pixel ----
    for (int p = tid; p < NSRC; p += 256) {
        const int r = p / SRC, c = p % SRC;
        const int gy = gy0 + r, gx = gx0 + c;
        unsigned bits = 0u;
        if ((unsigned)gy < HH && (unsigned)gx < WW) {
            const unsigned char* m = bmask + ((((size_t)n*HH + gy)*WW) + gx) * KK;
            bits = (m[0] ? 0u : 1u) | (m[1] ? 0u : 2u) | (m[2] ? 0u : 4u) | (m[3] ? 0u : 8u);
        }
        smask[p] = bits;
    }

    // ---- uniform lighting params (scalar-cached loads) ----
    const float Lx = lpos[0], Ly = lpos[1], Lz = lpos[2];
    const float Cx = cpos[0], Cy = cpos[1], Cz = cpos[2];
    const float ambR = lamb[0]*mamb[0], ambG = lamb[1]*mamb[1], ambB = lamb[2]*mamb[2];
    const float difR = ldif[0]*mdif[0], difG = ldif[1]*mdif[1], difB = ldif[2]*mdif[2];
    const float spcR = lspc[0]*mspc[0], spcG = lspc[1]*mspc[1], spcB = lspc[2]*mspc[2];

    // ---- stage 3: Phong shade 18x18x4 source layers into LDS (overlaps DMA) ----
    for (int i = tid; i < NSRC*KK; i += 256) {
        const int p = i >> 2, k = i & 3;
        const int r = p / SRC, c = p % SRC;
        const int gy = gy0 + r, gx = gx0 + c;
        if ((unsigned)gy < HH && (unsigned)gx < WW) {
            const size_t b3 = ((((((size_t)n*HH + gy)*WW) + gx)*KK) + k) * 3;
            float nx = normals[b3], ny = normals[b3+1], nz = normals[b3+2];
            float px = pcc[b3],     py = pcc[b3+1],     pz = pcc[b3+2];
            float inv = rsqrtf(nx*nx + ny*ny + nz*nz + 1e-8f); nx*=inv; ny*=inv; nz*=inv;
            float lx = Lx-px, ly = Ly-py, lz = Lz-pz;
            inv = rsqrtf(lx*lx + ly*ly + lz*lz + 1e-8f); lx*=inv; ly*=inv; lz*=inv;
            float vx = Cx-px, vy = Cy-py, vz = Cz-pz;
            inv = rsqrtf(vx*vx + vy*vy + vz*vz + 1e-8f); vx*=inv; vy*=inv; vz*=inv;
            const float ndl = nx*lx + ny*ly + nz*lz;
            const float rx = 2.f*ndl*nx - lx, ry = 2.f*ndl*ny - ly, rz = 2.f*ndl*nz - lz;
            const float rdv = fmaxf(rx*vx + ry*vy + rz*vz, 0.f);
            float s = rdv*rdv; s*=s; s*=s; s*=s; s*=s; s*=s;     // rdv^64
            const float dt = fmaxf(ndl, 0.f);
            srgb4[i] = make_float4(texels[b3  ]*(ambR + difR*dt) + spcR*s,
                                   texels[b3+1]*(ambG + difG*dt) + spcG*s,
                                   texels[b3+2]*(ambB + difB*dt) + spcB*s,
                                   0.f);
        }
    }

    wait_asynccnt0();          // drain async DMA issued by this wave
    __syncthreads();           // all LDS tiles visible

    // ---- stage 4: gather 9 directions x 4 layers, occlusion-bucketed blend ----
    const int oy = by*TILE + threadIdx.y, ox = bx*TILE + threadIdx.x;
    const int cp = (threadIdx.y + 1)*SRC + (threadIdx.x + 1);
    const float4 qd = sdep4[cp];
    const float cyf = oy + 0.5f, cxf = ox + 0.5f;
    // (1+eps) / sum_d exp(-|off_d|^2 / (2*sigma^2)),  2*sigma^2 = 0.5
    const float norm_c = 1.05f / (4.f*expf(-4.f) + 4.f*expf(-2.f) + 1.f);

    // Inside the mask guard alpha == 1, so the reference's alpha*w accumulator is
    // bitwise identical to the w accumulator -> track only (r,g,b,w) per bucket.
    float fg0=0,fg1=0,fg2=0,fgw=0;
    float sf0=0,sf1=0,sf2=0,sfw=0;
    float bg0=0,bg1=0,bg2=0,bgw=0;

    #pragma unroll
    for (int d = 0; d < 9; ++d) {
        const int dy = d/3 - 1, dx = d%3 - 1;              // compile-time after unroll
        const int sp = cp - dy*SRC - dx;                   // source pixel for this direction
        const float4 pd = sdep4[sp];
        // dq = |p_top - q_depth[k]|, first-occurrence argmin
        float dqm = fabsf(pd.x - qd.x); int lq = 0; float v;
        v = fabsf(pd.x - qd.y); if (v < dqm) { dqm = v; lq = 1; }
        v = fabsf(pd.x - qd.z); if (v < dqm) { dqm = v; lq = 2; }
        v = fabsf(pd.x - qd.w); if (v < dqm) { dqm = v; lq = 3; }
        // dp = |q_top - p_depth[k]|
        float dpm = fabsf(qd.x - pd.x); int lp = 0;
        v = fabsf(qd.x - pd.y); if (v < dpm) { dpm = v; lp = 1; }
        v = fabsf(qd.x - pd.z); if (v < dpm) { dpm = v; lp = 2; }
        v = fabsf(qd.x - pd.w); if (v < dpm) { dpm = v; lp = 3; }
        const int occ = (dqm <= dpm) ? lq : -lp;

        const unsigned am = smask[sp];
        if (am) {
            const float4 c01 = scrd4[sp*2 + 0];
            const float4 c23 = scrd4[sp*2 + 1];
            #pragma unroll
            for (int k = 0; k < KK; ++k) {
                if (am & (1u << k)) {
                    const float sy = (k < 2) ? (k == 0 ? c01.x : c01.z) : (k == 2 ? c23.x : c23.z);
                    const float sx = (k < 2) ? (k == 0 ? c01.y : c01.w) : (k == 2 ? c23.y : c23.w);
                    const float ddy = sy - cyf, ddx = sx - cxf;
                    const float w = expf(-2.f*(ddx*ddx + ddy*ddy)) * norm_c;
                    const float4 rgb = srgb4[sp*4 + k];
                    const float rr = rgb.x*w, gg = rgb.y*w, bb = rgb.z*w;
                    const int level = k + occ;
                    const float mFg = (level < 0)  ? 1.f : 0.f;
                    const float mSf = (level == 0) ? 1.f : 0.f;
                    const float mBg = 1.f - mFg - mSf;
                    fg0 += rr*mFg; fg1 += gg*mFg; fg2 += bb*mFg; fgw += w*mFg;
                    sf0 += rr*mSf; sf1 += gg*mSf; sf2 += bb*mSf; sfw += w*mSf;
                    bg0 += rr*mBg; bg1 += gg*mBg; bg2 += bb*mBg; bgw += w*mBg;
                }
            }
        }
    }

    // ---- stage 5: normalize buckets, composite fg over surf over bg over white ----
    float inv = 1.f / fmaxf(bgw, 1e-10f);
    float o0 = bg0*inv, o1 = bg1*inv, o2 = bg2*inv, o3 = bgw*inv;
    inv = 1.f / fmaxf(sfw, 1e-10f);
    { const float s0=sf0*inv, s1=sf1*inv, s2=sf2*inv, s3=sfw*inv;
      o0 = s0 + (1.f - s3)*o0; o1 = s1 + (1.f - s3)*o1;
      o2 = s2 + (1.f - s3)*o2; o3 = s3 + (1.f - s3)*o3; }
    inv = 1.f / fmaxf(fgw, 1e-10f);
    { const float f0=fg0*inv, f1=fg1*inv, f2=fg2*inv, f3=fgw*inv;
      o0 = f0 + (1.f - f3)*o0; o1 = f1 + (1.f - f3)*o1;
      o2 = f2 + (1.f - f3)*o2; o3 = f3 + (1.f - f3)*o3; }

    const float4 res = make_float4(o0 + (1.f - o3), o1 + (1.f - o3), o2 + (1.f - o3), o3);
    *reinterpret_cast<float4*>(out + ((((size_t)n*HH + oy)*WW) + ox) * 4) = res;
}

extern "C" void kernel_launch(void* const* d_in, const int* in_sizes, int n_in,
                              void* d_out, int out_size, void* d_ws, size_t ws_size,
                              hipStream_t stream) {
    (void)in_sizes; (void)n_in; (void)d_ws; (void)ws_size; (void)out_size;
    dim3 block(TILE, TILE, 1);
    dim3 grid(WW / TILE, HH / TILE, NN);
    splat_phong_kernel<<<grid, block, 0, stream>>>(
        (const float*)d_in[0],          // texels
        (const float*)d_in[1],          // pixel_normals
        (const float*)d_in[2],          // pixel_coords_cameras
        (const float*)d_in[3],          // pixel_coords_screen
        (const float*)d_in[4],          // q_depth
        (const unsigned char*)d_in[5],  // background_mask (bool bytes)
        (const float*)d_in[6],          // light_position
        (const float*)d_in[7],          // light_ambient
        (const float*)d_in[8],          // light_diffuse
        (const float*)d_in[9],          // light_specular
        (const float*)d_in[10],         // camera_position
        (const float*)d_in[11],         // mat_ambient
        (const float*)d_in[12],         // mat_diffuse
        (const float*)d_in[13],         // mat_specular
        (float*)d_out);
}